// Scene_2980707304172
// MI455X (gfx1250) — compile-verified
//
#include <hip/hip_runtime.h>

// Scene splat: out[c, y0+r, x0+col] += patch[n, c, r, col] for 2000 patches.
// Memory-bound scatter-add; uses gfx1250 async global->LDS staging (ASYNCcnt)
// + native f32 L2 atomics. No matmul structure -> WMMA intentionally not used.

constexpr int kC = 5;
constexpr int kH = 4096;
constexpr int kW = 4096;
constexpr int kPH = 64;
constexpr int kPW = 64;

__global__ void scene_zero_kernel(float4* __restrict__ out, long n4) {
  long i = (long)blockIdx.x * blockDim.x + threadIdx.x;
  if (i < n4) {
    out[i] = float4{0.f, 0.f, 0.f, 0.f};
  }
}

__global__ __launch_bounds__(256)
void scene_splat_kernel(const float* __restrict__ patches,
                        const int* __restrict__ positions,
                        float* __restrict__ out) {
  // One block per (source n, band c) plane: 64*64 = 4096 floats = 16 KB.
  __shared__ float tile[kPH * kPW];

  const int plane = blockIdx.x;           // plane = n*C + c
  const int n = plane / kC;
  const int c = plane - n * kC;
  const int y0 = positions[2 * n + 0];
  const int x0 = positions[2 * n + 1];

  const float* __restrict__ pbase = patches + (size_t)plane * (kPH * kPW);
  float* __restrict__ sbase =
      out + (size_t)c * kH * kW + (size_t)y0 * kW + (size_t)x0;

  const int t = threadIdx.x;  // 0..255 -> 8 wave32 per block

  // Stage the whole patch plane into LDS with async b128 copies.
  // Patch data is streamed exactly once -> non-temporal hint so it does not
  // evict the scene's atomic working set from the 192MB L2.
#pragma unroll
  for (int k = 0; k < 4; ++k) {
    const int idx4 = t + k * 256;                       // float4 index 0..1023
    unsigned lds_byte = (unsigned)(size_t)(const void*)&tile[idx4 * 4];
    unsigned gbyte = (unsigned)(idx4 * 16);             // byte offset in plane
    asm volatile("global_load_async_to_lds_b128 %0, %1, %2 th:TH_LOAD_NT"
                 :
                 : "v"(lds_byte), "v"(gbyte), "s"(pbase)
                 : "memory");
  }
  // Wait for the async engine to land all 16KB in LDS (ASYNCcnt -> 0).
  // Each thread only reads back the entries it staged, so no barrier needed.
  asm volatile("s_wait_asynccnt 0x0" ::: "memory");

  // Splat into the scene with native f32 atomics (overlapping patches race).
#pragma unroll
  for (int k = 0; k < 4; ++k) {
    const int idx4 = t + k * 256;
    const int row = idx4 >> 4;         // 16 float4 per 64-float patch row
    const int col = (idx4 & 15) << 2;
    const float4 v = *reinterpret_cast<const float4*>(&tile[idx4 * 4]);
    float* dst = sbase + (size_t)row * kW + col;
    __hip_atomic_fetch_add(dst + 0, v.x, __ATOMIC_RELAXED, __HIP_MEMORY_SCOPE_AGENT);
    __hip_atomic_fetch_add(dst + 1, v.y, __ATOMIC_RELAXED, __HIP_MEMORY_SCOPE_AGENT);
    __hip_atomic_fetch_add(dst + 2, v.z, __ATOMIC_RELAXED, __HIP_MEMORY_SCOPE_AGENT);
    __hip_atomic_fetch_add(dst + 3, v.w, __ATOMIC_RELAXED, __HIP_MEMORY_SCOPE_AGENT);
  }
}

extern "C" void kernel_launch(void* const* d_in, const int* in_sizes, int n_in,
                              void* d_out, int out_size, void* d_ws, size_t ws_size,
                              hipStream_t stream) {
  const float* patches = (const float*)d_in[0];    // (N, C, 64, 64) fp32
  const int* positions = (const int*)d_in[1];      // (N, 2) int32
  float* out = (float*)d_out;                      // (C, 4096, 4096) fp32

  // 1) Zero the scene with vectorized (b128) stores.
  const long n4 = (long)out_size / 4;              // out_size divisible by 4
  const int zblk = 256;
  const long zgrid = (n4 + zblk - 1) / zblk;
  scene_zero_kernel<<<(unsigned)zgrid, zblk, 0, stream>>>((float4*)out, n4);

  // 2) Splat all patches; one block per (n, c) plane.
  const int N = in_sizes[1] / 2;                   // positions is (N,2)
  scene_splat_kernel<<<N * kC, 256, 0, stream>>>(patches, positions, out);
}